// HyperMoment_47493748359216
// MI455X (gfx1250) — compile-verified
//
#include <hip/hip_runtime.h>

// ---------------------------------------------------------------------------
// Hyper-moment contraction, MI455X (gfx1250).
// Memory-bound streaming kernel: 253 MB moved, ~0.7 GFLOP -> HBM roofline
// (~11 us at 23.3 TB/s). CDNA5-specific path: async global<->LDS staging
// (ASYNCcnt-tracked GLOBAL_LOAD_ASYNC_TO_LDS / GLOBAL_STORE_ASYNC_FROM_LDS)
// with s_wait_asynccnt, plus NT hints on the fallback path.
// ---------------------------------------------------------------------------

#define NU       8
#define NATOMS   100000
#define TILE     128

#define OUT1_OFF 800000      // 8*100000
#define OUT2_OFF 3200000     // + 8*100000*3
#define OUT3_OFF 10400000    // + 8*100000*9

typedef float vf4 __attribute__((ext_vector_type(4)));
typedef int   vi4 __attribute__((ext_vector_type(4)));

#define AS_GLOBAL __attribute__((address_space(1)))
#define AS_LOCAL  __attribute__((address_space(3)))

// Two-step casts: AS cast on void* (proven to compile), then pointee cast.
#define GPTR4(p) ((AS_GLOBAL vi4*)(AS_GLOBAL void*)(p))
#define LPTR4(p) ((AS_LOCAL  vi4*)(AS_LOCAL  void*)(p))
#define GPTR1(p) ((AS_GLOBAL int*)(AS_GLOBAL void*)(p))
#define LPTR1(p) ((AS_LOCAL  int*)(AS_LOCAL  void*)(p))

#if defined(__HIP_DEVICE_COMPILE__)
#  if __has_builtin(__builtin_amdgcn_global_load_async_to_lds_b128) && \
      __has_builtin(__builtin_amdgcn_global_load_async_to_lds_b32)  && \
      __has_builtin(__builtin_amdgcn_global_store_async_from_lds_b128) && \
      __has_builtin(__builtin_amdgcn_global_store_async_from_lds_b32)  && \
      __has_builtin(__builtin_amdgcn_s_wait_asynccnt)
#    define HM_ASYNC 1
#  else
#    define HM_ASYNC 0
#    warning "gfx1250 async_to_lds builtins not found: plain LDS staging fallback"
#  endif
#else
#  define HM_ASYNC 0
#endif

// Coalesced global -> LDS staging (b128 main + b32 tail).
__device__ __forceinline__ void stage_in(float* lds, const float* g,
                                         int nFloats, int t)
{
  const int nv = nFloats >> 2;
#if HM_ASYNC
  for (int i = t; i < nv; i += TILE)
    __builtin_amdgcn_global_load_async_to_lds_b128(
        GPTR4(g + 4 * i), LPTR4(lds + 4 * i), 0, 0);
  for (int i = (nv << 2) + t; i < nFloats; i += TILE)
    __builtin_amdgcn_global_load_async_to_lds_b32(
        GPTR1(g + i), LPTR1(lds + i), 0, 0);
#else
  for (int i = t; i < nv; i += TILE)
    *(vf4*)(lds + 4 * i) = __builtin_nontemporal_load((const vf4*)(g + 4 * i));
  for (int i = (nv << 2) + t; i < nFloats; i += TILE)
    lds[i] = __builtin_nontemporal_load(g + i);
#endif
}

// Coalesced LDS -> global streaming store (b128 main + b32 tail).
__device__ __forceinline__ void stage_out(float* g, const float* lds,
                                          int nFloats, int t)
{
  const int nv = nFloats >> 2;
#if HM_ASYNC
  for (int i = t; i < nv; i += TILE)
    __builtin_amdgcn_global_store_async_from_lds_b128(
        GPTR4(g + 4 * i), LPTR4(lds + 4 * i), 0, 0);
  for (int i = (nv << 2) + t; i < nFloats; i += TILE)
    __builtin_amdgcn_global_store_async_from_lds_b32(
        GPTR1(g + i), LPTR1(lds + i), 0, 0);
  // completion covered by S_ENDPGM's implicit wait-idle
#else
  for (int i = t; i < nv; i += TILE) {
    vf4 val = *(const vf4*)(lds + 4 * i);
    __builtin_nontemporal_store(val, (vf4*)(g + 4 * i));
  }
  for (int i = (nv << 2) + t; i < nFloats; i += TILE)
    __builtin_nontemporal_store(lds[i], g + i);
#endif
}

__global__ __launch_bounds__(TILE) void hyper_moment_kernel(
    const float* __restrict__ m1, const float* __restrict__ m2,
    const float* __restrict__ m3,
    const float* __restrict__ W0, const float* __restrict__ W1,
    const float* __restrict__ W2, const float* __restrict__ W3,
    float* __restrict__ out)
{
  __shared__ alignas(16) float sI1[TILE * 3];
  __shared__ alignas(16) float sI2[TILE * 9];
  __shared__ alignas(16) float sI3[TILE * 27];
  __shared__ alignas(16) float sO0[TILE];
  __shared__ alignas(16) float sO1[TILE * 3];
  __shared__ alignas(16) float sO2[TILE * 9];
  __shared__ alignas(16) float sO3[TILE * 27];

  const int z  = blockIdx.y;
  const int a0 = blockIdx.x * TILE;
  const int t  = threadIdx.x;
  int n = NATOMS - a0;
  if (n > TILE) n = TILE;

  const size_t base = (size_t)z * NATOMS + (size_t)a0;

  stage_in(sI1, m1 + base * 3,  n * 3,  t);
  stage_in(sI2, m2 + base * 9,  n * 9,  t);
  stage_in(sI3, m3 + base * 27, n * 27, t);

#if HM_ASYNC
  __builtin_amdgcn_s_wait_asynccnt(0);
#endif
  __syncthreads();

  // Per-channel mixing weights: uniform within the block -> scalar loads.
  const float w00 = W0[z], w01 = W0[NU + z], w02 = W0[2 * NU + z];
  const float w10 = W1[z], w11 = W1[NU + z], w12 = W1[2 * NU + z];
  const float w20 = W2[z], w21 = W2[NU + z], w22 = W2[2 * NU + z],
              w23 = W2[3 * NU + z], w24 = W2[4 * NU + z];
  const float w30 = W3[z], w31 = W3[NU + z], w32 = W3[2 * NU + z];

  if (t < n) {
    float v[3], M[9], T[27];
#pragma unroll
    for (int i = 0; i < 3; ++i)  v[i] = sI1[t * 3 + i];
#pragma unroll
    for (int i = 0; i < 9; ++i)  M[i] = sI2[t * 9 + i];
#pragma unroll
    for (int i = 0; i < 27; ++i) T[i] = sI3[t * 27 + i];

    // ---- rank 0: v.v, M:M, T:T ----
    float s1 = 0.f, s2 = 0.f, s3 = 0.f;
#pragma unroll
    for (int i = 0; i < 3;  ++i) s1 += v[i] * v[i];
#pragma unroll
    for (int i = 0; i < 9;  ++i) s2 += M[i] * M[i];
#pragma unroll
    for (int i = 0; i < 27; ++i) s3 += T[i] * T[i];
    sO0[t] = w00 * s1 + w01 * s2 + w02 * s3;

    // ---- rank 1 ----
#pragma unroll
    for (int c = 0; c < 3; ++c) {
      // zab,zacb->zac : sum_b v[b] * M[c,b]
      float h1 = M[c * 3 + 0] * v[0] + M[c * 3 + 1] * v[1] + M[c * 3 + 2] * v[2];
      // zabc,zadbc->zad : sum_{b,c2} M[b,c2] * T[d, b, c2]  (flattened)
      float h2 = 0.f;
#pragma unroll
      for (int q = 0; q < 9; ++q) h2 += M[q] * T[c * 9 + q];
      sO1[t * 3 + c] = w10 * v[c] + w11 * h1 + w12 * h2;
    }

    // ---- rank 2 (output index i,j) ----
#pragma unroll
    for (int i = 0; i < 3; ++i) {
#pragma unroll
      for (int j = 0; j < 3; ++j) {
        // zab,zac->zabc : outer product
        float p1 = v[i] * v[j];
        // zab,zacdb->zacd : sum_b v[b] * T[i, j, b]
        float p2 = v[0] * T[i * 9 + j * 3 + 0] +
                   v[1] * T[i * 9 + j * 3 + 1] +
                   v[2] * T[i * 9 + j * 3 + 2];
        // zabc,zadc->zabd : sum_c M[i,c] * M[j,c]
        float p3 = M[i * 3 + 0] * M[j * 3 + 0] +
                   M[i * 3 + 1] * M[j * 3 + 1] +
                   M[i * 3 + 2] * M[j * 3 + 2];
        // zabcd,zaecd->zabe : sum_{c,d} T[i,c,d] * T[j,c,d]
        float p4 = 0.f;
#pragma unroll
        for (int q = 0; q < 9; ++q) p4 += T[i * 9 + q] * T[j * 9 + q];
        sO2[t * 9 + i * 3 + j] =
            w20 * M[i * 3 + j] + w21 * p1 + w22 * p2 + w23 * p3 + w24 * p4;
      }
    }

    // ---- rank 3 (output index i, jk=j*3+k) ----
#pragma unroll
    for (int i = 0; i < 3; ++i) {
#pragma unroll
      for (int jk = 0; jk < 9; ++jk) {
        // zabc,zadec->zabde : sum_c M[i,c] * T[j,k,c]; T[j*9+k*3+c]=T[jk*3+c]
        float q2 = M[i * 3 + 0] * T[jk * 3 + 0] +
                   M[i * 3 + 1] * T[jk * 3 + 1] +
                   M[i * 3 + 2] * T[jk * 3 + 2];
        sO3[t * 27 + i * 9 + jk] =
            w30 * T[i * 9 + jk] + w31 * v[i] * M[jk] + w32 * q2;
      }
    }
  }
  __syncthreads();

  stage_out(out + base,                 sO0, n,      t);
  stage_out(out + OUT1_OFF + base * 3,  sO1, n * 3,  t);
  stage_out(out + OUT2_OFF + base * 9,  sO2, n * 9,  t);
  stage_out(out + OUT3_OFF + base * 27, sO3, n * 27, t);
}

extern "C" void kernel_launch(void* const* d_in, const int* in_sizes, int n_in,
                              void* d_out, int out_size, void* d_ws,
                              size_t ws_size, hipStream_t stream)
{
  (void)in_sizes; (void)n_in; (void)out_size; (void)d_ws; (void)ws_size;
  const float* m1 = (const float*)d_in[0];
  const float* m2 = (const float*)d_in[1];
  const float* m3 = (const float*)d_in[2];
  const float* W0 = (const float*)d_in[3];
  const float* W1 = (const float*)d_in[4];
  const float* W2 = (const float*)d_in[5];
  const float* W3 = (const float*)d_in[6];
  float* out = (float*)d_out;

  dim3 grid((NATOMS + TILE - 1) / TILE, NU, 1);
  hyper_moment_kernel<<<grid, dim3(TILE, 1, 1), 0, stream>>>(
      m1, m2, m3, W0, W1, W2, W3, out);
}